// RoIPoolingLayer_5815385718818
// MI455X (gfx1250) — compile-verified
//
#include <hip/hip_runtime.h>
#include <hip/hip_bf16.h>

// RoI max pooling, matching the JAX reference:
//   features: (B=4, C=256, H=56, W=56) fp32
//   rois:     (N=256, 5) fp32  [b, x1, y1, x2, y2]
//   out:      (N, C, 7, 7) fp32
//
// MI455X design (compile-only reasoning):
//  - No MAC structure -> WMMA not applicable. Features (12.8MB) live in the
//    192MB L2; unique-byte roofline at 23.3TB/s is ~1us. Bottleneck = gather
//    issue rate / latency hiding, so the goal is batched load clauses.
//  - 1 block per RoI, 1 thread per channel: all bin bounds wave-uniform ->
//    scalar-FP RoI decode + SALU control (confirmed in earlier asm), zero
//    divergence.
//  - Inner w-window is FIXED at 9 with -inf masking (identical to the
//    reference's MBW=9 masked-window formulation): 9 loads issue as one
//    clause with one wait instead of load->wait->max per element (round-2
//    histogram showed 361 waits for 457 loads).
//  - Wave-uniform scalar branch picks an immediate-offset fast path when the
//    window needs no right-edge clamping (the common case).
//  - max3 trees shorten the post-wait reduction (v_max3_num_f32).
//  - gfx1250 global_prefetch_b8 warms each lane's channel-plane RoI origin.

#define POOLED_H 7
#define POOLED_W 7
#define FEAT_C   256
#define FEAT_H   56
#define FEAT_W   56
#define MBW      9           // ceil(W/7)+1, max bin width (= reference MBW)
#define SPATIAL_SCALE 0.0625f

#define NEG_INF (-__builtin_huge_valf())

__global__ __launch_bounds__(FEAT_C) void roi_pool_kernel(
    const float* __restrict__ features,   // (B, C, H, W)
    const float* __restrict__ rois,       // (N, 5)
    float* __restrict__ out)              // (N, C, 7, 7)
{
    const int n = blockIdx.x;      // RoI index
    const int c = threadIdx.x;     // channel index (256 threads)

    // Block-uniform RoI parameters -> scalar loads + scalar float decode.
    const float* r = rois + (size_t)n * 5;
    const int b  = (int)r[0];
    const int x1 = (int)(r[1] * SPATIAL_SCALE);
    const int y1 = (int)(r[2] * SPATIAL_SCALE);
    const int x2 = (int)(r[3] * SPATIAL_SCALE);
    const int y2 = (int)(r[4] * SPATIAL_SCALE);

    const int roi_h = y2 - y1 + 1;   // >= 1 by construction
    const int roi_w = x2 - x1 + 1;   // >= 1 by construction

    // Column bin bounds independent of ph: compute once (wave-uniform,
    // folds to scalars under full unroll).
    int ws[POOLED_W], wid[POOLED_W];
    #pragma unroll
    for (int pw = 0; pw < POOLED_W; ++pw) {
        const int s = x1 + (pw * roi_w) / POOLED_W;
        const int e = x1 + ((pw + 1) * roi_w + POOLED_W - 1) / POOLED_W;
        ws[pw]  = s;
        wid[pw] = e - s;             // 1..9 valid elements of the 9-window
    }

    // Per-lane channel plane base.
    const float* __restrict__ fc =
        features + ((size_t)b * FEAT_C + c) * (FEAT_H * FEAT_W);
    float* __restrict__ oc =
        out + ((size_t)n * FEAT_C + c) * (POOLED_H * POOLED_W);

    // gfx1250 prefetch: warm the RoI origin of this lane's channel plane.
    {
        const int h0 = min(max(y1, 0), FEAT_H - 1);
        const int w0 = min(max(x1, 0), FEAT_W - 1);
        __builtin_prefetch(fc + h0 * FEAT_W + w0, 0, 1);
    }

    #pragma unroll
    for (int ph = 0; ph < POOLED_H; ++ph) {
        const int hs = y1 + (ph * roi_h) / POOLED_H;
        const int he = y1 + ((ph + 1) * roi_h + POOLED_H - 1) / POOLED_H;

        // 7 independent accumulators -> overlapping load/max chains.
        float m[POOLED_W];
        #pragma unroll
        for (int pw = 0; pw < POOLED_W; ++pw) m[pw] = NEG_INF;

        for (int h = hs; h < he; ++h) {
            const int hc = min(max(h, 0), FEAT_H - 1);   // reference clip
            const float* __restrict__ row = fc + hc * FEAT_W;

            #pragma unroll
            for (int pw = 0; pw < POOLED_W; ++pw) {
                const int s = ws[pw];
                const int k = wid[pw];
                float v[MBW];

                if (s + (MBW - 1) < FEAT_W) {
                    // Fast path (wave-uniform branch): immediate offsets,
                    // 9 loads in one clause, no per-load address math.
                    const float* __restrict__ p = row + s;
                    #pragma unroll
                    for (int i = 0; i < MBW; ++i) v[i] = p[i];
                } else {
                    // Right-edge path: clip indices like the reference.
                    #pragma unroll
                    for (int i = 0; i < MBW; ++i)
                        v[i] = row[min(s + i, FEAT_W - 1)];
                }

                // Mask invalid window slots with -inf (scalar condition ->
                // s_cmp + v_cndmask), exactly as the reference does.
                #pragma unroll
                for (int i = 1; i < MBW; ++i)
                    v[i] = (i < k) ? v[i] : NEG_INF;   // slot 0 always valid

                // max3-tree reduction: 4x v_max3 + 1x v_max.
                const float t0 = fmaxf(fmaxf(v[0], v[1]), v[2]);
                const float t1 = fmaxf(fmaxf(v[3], v[4]), v[5]);
                const float t2 = fmaxf(fmaxf(v[6], v[7]), v[8]);
                m[pw] = fmaxf(m[pw], fmaxf(fmaxf(t0, t1), t2));
            }
        }

        #pragma unroll
        for (int pw = 0; pw < POOLED_W; ++pw)
            oc[ph * POOLED_W + pw] = m[pw];
    }
}

extern "C" void kernel_launch(void* const* d_in, const int* in_sizes, int n_in,
                              void* d_out, int out_size, void* d_ws, size_t ws_size,
                              hipStream_t stream) {
    const float* features = (const float*)d_in[0];
    const float* rois     = (const float*)d_in[1];
    float*       out      = (float*)d_out;

    const int N = in_sizes[1] / 5;   // number of RoIs (256)

    roi_pool_kernel<<<dim3(N), dim3(FEAT_C), 0, stream>>>(features, rois, out);
}